// L3_9474697855326
// MI455X (gfx1250) — compile-verified
//
#include <hip/hip_runtime.h>

// ---------------------------------------------------------------------------
// Fused Kronecker-factorized linear for MI455X (gfx1250, wave32).
//   y[b] = x[b] (32x32x32) contracted on each factor axis by W1, W2, W3.
// HBM traffic = read x once + write y once (512 MB) -> ~22 us floor @23.3TB/s.
// One workgroup (32 waves) owns one batch element at a time; all three
// 32-point contractions run in LDS with V_WMMA_F32_16X16X4_F32.
// The 128 KB input tile is DMA'd HBM->LDS by the Tensor Data Mover
// (tensor_load_to_lds), double-buffered so the next tile streams in while
// the current one is computed (TENSORcnt + barrier handshake).
// ---------------------------------------------------------------------------

typedef float v2f __attribute__((ext_vector_type(2)));
typedef float v8f __attribute__((ext_vector_type(8)));
typedef unsigned int v4u __attribute__((ext_vector_type(4)));
typedef int v4i __attribute__((ext_vector_type(4)));
typedef int v8i __attribute__((ext_vector_type(8)));

#if defined(__has_builtin)
#if __has_builtin(__builtin_amdgcn_wmma_f32_16x16x4_f32)
#define USE_WMMA_F32 1
#endif
#if __has_builtin(__builtin_amdgcn_tensor_load_to_lds) && \
    __has_builtin(__builtin_amdgcn_s_wait_tensorcnt)
#define USE_TDM 1
#endif
#endif
#ifndef USE_WMMA_F32
#define USE_WMMA_F32 0
#endif
#ifndef USE_TDM
#define USE_TDM 0
#endif

#define NBATCH 2048
#define GRIDX  512
#define ITERS  (NBATCH / GRIDX)

// LDS layout (floats): W1[1024] W2[1024] W3[1024] buf0[33792] buf1[33792]
#define BUF_FLOATS  33792
#define BUF_OFF(c)  (3072 + (c) * BUF_FLOATS)            // float index
#define SMEM_FLOATS (3072 + 2 * BUF_FLOATS)              // 70656 -> 276 KB

// ---------------- Tensor Data Mover: one 32x1024-f32 tile HBM -> LDS -------
#if USE_TDM
__device__ __forceinline__ void tdm_load_tile(const float* gsrc, unsigned lds_byte_off)
{
    const unsigned long long ga = (unsigned long long)(__SIZE_TYPE__)gsrc;
    // D# group 0: count=1 | lds_addr | global_addr[56:0] | type=2 (ISA 8.3)
    v4u g0;
    g0.x = 1u;                                            // count=1, user D#
    g0.y = lds_byte_off;                                  // lds_addr (bytes)
    g0.z = (unsigned)ga;                                  // global_addr[31:0]
    g0.w = ((unsigned)(ga >> 32) & 0x01FFFFFFu) | (2u << 30); // [56:32]|type=2
    // D# group 1: data_size=4B, tensor 1024x32, tile 1024x32, stride0=1024
    v8i g1;
    g1[0] = (int)(2u << 16);          // workgroup_mask=0, data_size=2 (4B)
    g1[1] = (int)(1024u << 16);       // tensor_dim0.lo16 = 1024
    g1[2] = (int)(32u << 16);         // tensor_dim0.hi16=0 | tensor_dim1.lo16=32
    g1[3] = (int)(1024u << 16);       // tensor_dim1.hi16=0 | tile_dim0=1024
    g1[4] = 32;                       // tile_dim1=32 | tile_dim2=0
    g1[5] = 1024;                     // tensor_dim0_stride.lo32=1024
    g1[6] = 0;                        // stride0.hi16=0 | stride1.lo16=0
    g1[7] = 0;
    const v4i z4 = {0, 0, 0, 0};      // groups 2/3 unused (2D tensor)
    const v8i z8 = {0, 0, 0, 0, 0, 0, 0, 0};
    // clang-23 / therock toolchain: 6-argument form (extra int32x8 group).
    __builtin_amdgcn_tensor_load_to_lds(g0, g1, z4, z4, z8, 0);
}
#endif

// ------------- one 16x16 += 16x4 * 4x16 WMMA step from LDS -----------------
// A element [m][k] = Ab[m*a_ms + k*a_ks];  B element [k][n] = Bb[k*b_ks + n].
__device__ __forceinline__ v8f wmma_k4(const float* __restrict__ Ab, int a_ms, int a_ks,
                                       const float* __restrict__ Bb, int b_ks,
                                       v8f acc, int lane)
{
    const int half = lane >> 4;
    const int l    = lane & 15;
#if USE_WMMA_F32
    const int kA = half << 1;         // hi-half lanes carry K=2,3
    v2f a, b;
    a.x = Ab[l * a_ms + kA * a_ks];
    a.y = Ab[l * a_ms + (kA + 1) * a_ks];
    b.x = Bb[kA * b_ks + l];
    b.y = Bb[(kA + 1) * b_ks + l];
    acc = __builtin_amdgcn_wmma_f32_16x16x4_f32(false, a, false, b,
                                                (short)0, acc, false, false);
#else
#pragma unroll
    for (int v = 0; v < 8; ++v) {
        const int m = v + half * 8;
        float s = acc[v];
#pragma unroll
        for (int k = 0; k < 4; ++k)
            s = __builtin_fmaf(Ab[m * a_ms + k * a_ks], Bb[k * b_ks + l], s);
        acc[v] = s;
    }
#endif
    return acc;
}

__global__ __launch_bounds__(1024)
void kron3_fused_kernel(const float* __restrict__ x,
                        const float* __restrict__ W1,
                        const float* __restrict__ W2,
                        const float* __restrict__ W3,
                        float* __restrict__ out)
{
    extern __shared__ float smem[];
    float* lW1 = smem;
    float* lW2 = smem + 1024;
    float* lW3 = smem + 2048;

    const int tid  = threadIdx.x;
    const int lane = tid & 31;
    const int wave = tid >> 5;
    const int half = lane >> 4;
    const int l    = lane & 15;

    // weights: 12 KB, loaded once per block (hot in L2 across all blocks)
    lW1[tid] = W1[tid];
    lW2[tid] = W2[tid];
    lW3[tid] = W3[tid];

    int b = blockIdx.x;
#if USE_TDM
    if (wave == 0)                                  // kick first tile DMA
        tdm_load_tile(x + (size_t)b * 32768, BUF_OFF(0) * 4u);
#endif

    int cur = 0;
    for (int iter = 0; iter < ITERS; ++iter, b += GRIDX) {
        float* buf = smem + BUF_OFF(cur);

#if USE_TDM
        if (wave == 0) __builtin_amdgcn_s_wait_tensorcnt(0); // tile b arrived
        __syncthreads();                 // release all waves; prior iter done
        const int bn = b + GRIDX;        // prefetch next tile during compute
        if (bn < NBATCH && wave == 0)
            tdm_load_tile(x + (size_t)bn * 32768, BUF_OFF(cur ^ 1) * 4u);
#else
        __syncthreads();                 // prior iteration finished with buf
        {
            const float4* __restrict__ xg = (const float4*)(x + (size_t)b * 32768);
            float4* __restrict__ b4 = (float4*)buf;
#pragma unroll
            for (int i = 0; i < 8; ++i) b4[tid + i * 1024] = xg[tid + i * 1024];
        }
        __syncthreads();
#endif

        v8f acc[4];

        // ============ step 1: Y1 = W1^T (32x32) @ X (32x1024) =============
#pragma unroll
        for (int s = 0; s < 4; ++s) {
            const int t  = s * 32 + wave;
            const int m0 = (t & 1) * 16;
            const int n0 = (t >> 1) * 16;
            v8f c = {0.f, 0.f, 0.f, 0.f, 0.f, 0.f, 0.f, 0.f};
            const float* Ab = lW1 + m0;            // A[m][k] = W1[k][m0+m]
            const float* Bb = buf + n0;            // B[k][n] = X[k][n0+n]
#pragma unroll
            for (int kk = 0; kk < 8; ++kk)
                c = wmma_k4(Ab + kk * 4 * 32, 1, 32, Bb + kk * 4 * 1024, 1024, c, lane);
            acc[s] = c;
        }
        __syncthreads();                           // all reads of X retired
#pragma unroll
        for (int s = 0; s < 4; ++s) {
            const int t  = s * 32 + wave;
            const int m0 = (t & 1) * 16;
            const int n0 = (t >> 1) * 16;
            float* D = buf + m0 * 1024 + n0;       // buf[i*1024 + q*32 + r]
#pragma unroll
            for (int v = 0; v < 8; ++v) D[(v + half * 8) * 1024 + l] = acc[s][v];
        }
        __syncthreads();

        // ======= step 2: per i, Y2_i = W2^T (32x32) @ Y1_i (32x32) ========
#pragma unroll
        for (int s = 0; s < 4; ++s) {
            const int t   = s * 32 + wave;
            const int i   = t >> 2;
            const int sub = t & 3;
            const int j0  = (sub & 1) * 16;
            const int r0  = (sub >> 1) * 16;
            v8f c = {0.f, 0.f, 0.f, 0.f, 0.f, 0.f, 0.f, 0.f};
            const float* Ab = lW2 + j0;                // A[m][k] = W2[k][j0+m]
            const float* Bb = buf + i * 1024 + r0;     // B[k][n] = Y1[i][k][r0+n]
#pragma unroll
            for (int kk = 0; kk < 8; ++kk)
                c = wmma_k4(Ab + kk * 4 * 32, 1, 32, Bb + kk * 4 * 32, 32, c, lane);
            acc[s] = c;
        }
        __syncthreads();                           // all reads of Y1 retired
        // Y2 written with r-stride 33 (odd) -> step-3 A loads are LDS
        // bank-conflict-free across all 64 banks.
#pragma unroll
        for (int s = 0; s < 4; ++s) {
            const int t   = s * 32 + wave;
            const int i   = t >> 2;
            const int sub = t & 3;
            const int j0  = (sub & 1) * 16;
            const int r0  = (sub >> 1) * 16;
            float* D = buf + i * 1056 + j0 * 33 + r0;
#pragma unroll
            for (int v = 0; v < 8; ++v) D[(v + half * 8) * 33 + l] = acc[s][v];
        }
        __syncthreads();

        // ==== step 3: Y = Y2 (1024x32) @ W3 (32x32), streamed NT to HBM ===
        float* __restrict__ og = out + (size_t)b * 32768;
#pragma unroll
        for (int s = 0; s < 4; ++s) {
            const int t  = s * 32 + wave;
            const int m0 = (t >> 1) * 16;          // ij rows m0..m0+15 share i
            const int n0 = (t & 1) * 16;
            v8f c = {0.f, 0.f, 0.f, 0.f, 0.f, 0.f, 0.f, 0.f};
            const float* Ab = buf + (m0 >> 5) * 1056 + (m0 & 31) * 33;
            const float* Bb = lW3 + n0;            // B[k][n] = W3[k][n0+n]
#pragma unroll
            for (int kk = 0; kk < 8; ++kk)
                c = wmma_k4(Ab + kk * 4, 33, 1, Bb + kk * 4 * 32, 32, c, lane);
            float* D = og + m0 * 32 + n0;          // 2 full 64B segments/store
#pragma unroll
            for (int v = 0; v < 8; ++v)
                __builtin_nontemporal_store(c[v], &D[(v + half * 8) * 32 + l]);
        }

        cur ^= 1;
    }
}

extern "C" void kernel_launch(void* const* d_in, const int* in_sizes, int n_in,
                              void* d_out, int out_size, void* d_ws, size_t ws_size,
                              hipStream_t stream) {
    (void)in_sizes; (void)n_in; (void)d_ws; (void)ws_size; (void)out_size;
    const float* x  = (const float*)d_in[0];
    const float* W1 = (const float*)d_in[1];
    const float* W2 = (const float*)d_in[2];
    const float* W3 = (const float*)d_in[3];
    float* out = (float*)d_out;

    constexpr size_t SMEM_BYTES = SMEM_FLOATS * sizeof(float);   // 276 KB
    (void)hipFuncSetAttribute((const void*)kron3_fused_kernel,
                              hipFuncAttributeMaxDynamicSharedMemorySize,
                              (int)SMEM_BYTES);
    kron3_fused_kernel<<<GRIDX, 1024, SMEM_BYTES, stream>>>(x, W1, W2, W3, out);
}